// LowRankSNN_61804579389751
// MI455X (gfx1250) — compile-verified
//
#include <hip/hip_runtime.h>
#include <math.h>

#define N_E 3200
#define N_I 800
#define NN  4000
#define TT  512

typedef float v2f __attribute__((ext_vector_type(2)));
typedef float v8f __attribute__((ext_vector_type(8)));

// ---------------------------------------------------------------------------
// Init: zero workspace state (phase, spk ping, gA, gB, spk pong = 5*NN floats)
// and the reference's zero boundary columns + readout region in d_out.
// ---------------------------------------------------------------------------
__global__ __launch_bounds__(256) void snn_init(float* __restrict__ ws,
                                                float* __restrict__ out) {
    const size_t NT = (size_t)NN * TT;
    int idx = blockIdx.x * 256 + threadIdx.x;
    if (idx < 5 * NN) ws[idx] = 0.0f;
    if (idx < NN) {
        out[(size_t)idx * TT + (TT - 1)] = 0.0f;   // V[:, T-1]   = 0
        out[NT     + (size_t)idx * TT]   = 0.0f;   // phase[:, 0] = 0
        out[2 * NT + (size_t)idx * TT]   = 0.0f;   // spk[:, 0]   = 0
        out[3 * NT + (size_t)idx * TT]   = 0.0f;   // I_syn[:, 0] = 0
    }
    if (idx < TT) out[4 * NT + idx] = 0.0f;        // readout
}

// ---------------------------------------------------------------------------
// One simulation step. Block b owns rows [16b, 16b+16). 8 waves split K.
// fp32 WMMA matvec: D(16x16) = A(16x4) x B(4x16) with B's 16 columns all
// replicated with the spike vector, so every D column equals conn_tile@spk.
// A layout (ISA 7.12.2, 32-bit A 16x4): lane m=l&15 is row M=m,
// VGPR v holds K = v + 2*(l>>4)  -> per-lane contiguous float2 load.
// C/D layout: lane 0 VGPRs 0..7 = rows 0..7, lane 16 = rows 8..15.
// ---------------------------------------------------------------------------
__global__ __launch_bounds__(256) void snn_step(
    const float* __restrict__ conn, const float* __restrict__ Input,
    const int* __restrict__ dtp, float* __restrict__ ws,
    float* __restrict__ out, int t) {

    __shared__ float sE[8][16];
    __shared__ float sI[8][16];

    const int tid  = threadIdx.x;
    const int wave = tid >> 5;
    const int lane = tid & 31;
    const int m    = lane & 15;
    const int h    = lane >> 4;
    const int r0   = blockIdx.x * 16;

    // ws slots: [0]=phase, [1]=spk buf0, [2]=gA, [3]=gB, [4]=spk buf1
    float* ph     = ws;
    float* gA     = ws + 2 * NN;   // gEE for E rows, gIE for I rows
    float* gB     = ws + 3 * NN;   // gEI for E rows, gII for I rows
    float* spk_rd = (((t - 1) & 1) == 0) ? (ws + NN) : (ws + 4 * NN);
    float* spk_wr = ((t & 1) == 0)       ? (ws + NN) : (ws + 4 * NN);

    const float* arow = conn + (size_t)(r0 + m) * NN + 2 * h;
    const float* bvec = spk_rd + 2 * h;

    v8f cE = {};   // accumulates over E columns (k < 3200)
    v8f cI = {};   // accumulates over I columns (k >= 3200)

    // E columns: 800 K-quads, 100 per wave, as 10 groups x 10 immediate-offset
    // quads so address math amortizes and loads clause/pipeline deeply.
    {
        const float* ap = arow + 4 * (100 * wave);
        const float* bp = bvec + 4 * (100 * wave);
        for (int g = 0; g < 10; ++g) {
#pragma unroll
            for (int u = 0; u < 10; ++u) {
                v2f a = *(const v2f*)(ap + 4 * u);
                v2f b = *(const v2f*)(bp + 4 * u);
                cE = __builtin_amdgcn_wmma_f32_16x16x4_f32(
                        false, a, false, b, (short)0, cE, false, false);
            }
            ap += 40; bp += 40;
        }
    }
    // I columns: 200 K-quads, 25 per wave, as 5 groups x 5 quads.
    {
        const float* ap = arow + 4 * (800 + 25 * wave);
        const float* bp = bvec + 4 * (800 + 25 * wave);
        for (int g = 0; g < 5; ++g) {
#pragma unroll
            for (int u = 0; u < 5; ++u) {
                v2f a = *(const v2f*)(ap + 4 * u);
                v2f b = *(const v2f*)(bp + 4 * u);
                cI = __builtin_amdgcn_wmma_f32_16x16x4_f32(
                        false, a, false, b, (short)0, cI, false, false);
            }
            ap += 20; bp += 20;
        }
    }

    // lanes 0 / 16 hold rows r0+0..7 / r0+8..15 in their 8 accumulators
    if (lane == 0) {
#pragma unroll
        for (int j = 0; j < 8; ++j) { sE[wave][j] = cE[j]; sI[wave][j] = cI[j]; }
    } else if (lane == 16) {
#pragma unroll
        for (int j = 0; j < 8; ++j) { sE[wave][8 + j] = cE[j]; sI[wave][8 + j] = cI[j]; }
    }
    __syncthreads();

    if (tid < 16) {
        float dE = 0.0f, dI = 0.0f;
#pragma unroll
        for (int w = 0; w < 8; ++w) { dE += sE[w][tid]; dI += sI[w][tid]; }

        const int i  = r0 + tid;
        const int iv = *dtp;
        const float dt = (iv > 0 && iv < 100000) ? (float)iv : __int_as_float(iv);

        const bool  isE = (i < N_E);
        const float gpa = isE ? 0.004069f : 0.003276f;   // G_P[0] / G_P[2]
        const float gpb = isE ? 0.02672f  : 0.02138f;    // G_P[1] / G_P[3]
        const float gL  = isE ? 0.08f     : 0.1f;

        // g_X = g_X + dt*(-g_X/tau + gp*d/dt)  ->  g_X + dt*(-g_X/tau) + gp*d
        float ga = gA[i];
        float gb = gB[i];
        ga = ga + dt * (-ga * 0.5f) + gpa * dE;   // tau_E = 2
        gb = gb + dt * (-gb * 0.2f) + gpb * dI;   // tau_I = 5

        const float phv  = ph[i];
        const float V    = -58.5f + 3.5f * __tanf(0.5f * phv);   // theta2V
        const float Isyn = -ga * V - gb * (V + 70.0f);           // REV_E=0, REV_I=-70
        const float inp  = Input[(size_t)i * TT + t];
        const float cosp = __cosf(phv);
        float pn = phv + dt * (-gL * cosp +
                   0.28571428571428570f * (1.0f + cosp) * (Isyn + inp)); // C0=2/7
        const float spike = (pn >= 3.14159265358979323f) ? 1.0f : 0.0f;
        pn -= 6.28318530717958648f * spike;

        gA[i]     = ga;
        gB[i]     = gb;
        ph[i]     = pn;
        spk_wr[i] = spike;

        const size_t NT = (size_t)NN * TT;
        out[(size_t)i * TT + (t - 1)]     = V;      // V[:, t-1]
        out[NT     + (size_t)i * TT + t]  = pn;     // phase[:, t]
        out[2 * NT + (size_t)i * TT + t]  = spike;  // spk[:, t]
        out[3 * NT + (size_t)i * TT + t]  = Isyn;   // I_syn[:, t]
    }
}

// ---------------------------------------------------------------------------
// readout[t] = W_out @ spk[:, t]  (deterministic LDS tree reduction)
// ---------------------------------------------------------------------------
__global__ __launch_bounds__(128) void snn_readout(const float* __restrict__ Wout,
                                                   float* __restrict__ out) {
    __shared__ float red[128];
    const int t = blockIdx.x;
    const size_t NT = (size_t)NN * TT;
    const float* spk = out + 2 * NT;
    float acc = 0.0f;
    for (int i = threadIdx.x; i < NN; i += 128)
        acc += Wout[i] * spk[(size_t)i * TT + t];
    red[threadIdx.x] = acc;
    __syncthreads();
    for (int s = 64; s > 0; s >>= 1) {
        if (threadIdx.x < s) red[threadIdx.x] += red[threadIdx.x + s];
        __syncthreads();
    }
    if (threadIdx.x == 0) out[4 * NT + t] = red[0];
}

// ---------------------------------------------------------------------------
extern "C" void kernel_launch(void* const* d_in, const int* in_sizes, int n_in,
                              void* d_out, int out_size, void* d_ws, size_t ws_size,
                              hipStream_t stream) {
    const int*   dtp   = (const int*)d_in[0];      // dt (python scalar)
    const float* Input = (const float*)d_in[1];    // (N, T) fp32 row-major
    const float* conn  = (const float*)d_in[2];    // (N, N) fp32 row-major
    const float* Wout  = (const float*)d_in[3];    // (N,)   fp32
    float* out = (float*)d_out;                    // [V | phase | spk | Isyn | readout]
    float* ws  = (float*)d_ws;                     // 5*NN floats of state

    snn_init<<<(5 * NN + 255) / 256, 256, 0, stream>>>(ws, out);
    for (int t = 1; t < TT; ++t)
        snn_step<<<NN / 16, 256, 0, stream>>>(conn, Input, dtp, ws, out, t);
    snn_readout<<<TT, 128, 0, stream>>>(Wout, out);
}